// TransNet_83133386981998
// MI455X (gfx1250) — compile-verified
//
#include <hip/hip_runtime.h>
#include <hip/hip_bf16.h>

typedef float v2f __attribute__((ext_vector_type(2)));
typedef float v8f __attribute__((ext_vector_type(8)));

#define EPS_LN 1e-5f

__device__ __forceinline__ float sigmoidf_(float t) {
    return 1.0f / (1.0f + __expf(-t));
}

// ---------------------------------------------------------------------------
// FP32 WMMA GEMM: C[n,KOUT] = A[n,128] @ W[128,KOUT] + bias.
// One wave computes a 16 x KOUT row-strip: A fragment loaded once per K-step,
// reused across COLT column tiles (COLT back-to-back v_wmma_f32_16x16x4_f32).
// ---------------------------------------------------------------------------
template <int KIN, int COLT>
__global__ void gemm_wmma_f32(const float* __restrict__ A,
                              const float* __restrict__ Wt,
                              const float* __restrict__ bias,
                              float* __restrict__ C, int nRowTiles) {
    int wave = (blockIdx.x * blockDim.x + threadIdx.x) >> 5;
    if (wave >= nRowTiles) return;            // wave-uniform: EXEC stays all-1s
    constexpr int KOUT = COLT * 16;
    int m0   = wave << 4;
    int lane = threadIdx.x & 31;
    int half = lane >> 4;                     // selects K sub-pair {0,1} vs {2,3}
    int l16  = lane & 15;

    const float* __restrict__ Arow = A + (size_t)(m0 + l16) * KIN;
    const v8f vzero = {0.f, 0.f, 0.f, 0.f, 0.f, 0.f, 0.f, 0.f};
    v8f acc[COLT];
#pragma unroll
    for (int t = 0; t < COLT; ++t) acc[t] = vzero;

    for (int k0 = 0; k0 < KIN; k0 += 4) {
        int ka = k0 + 2 * half;
        v2f a;
        a.x = Arow[ka];
        a.y = Arow[ka + 1];
        const float* __restrict__ w0 = Wt + (size_t)ka * KOUT + l16;
        const float* __restrict__ w1 = w0 + KOUT;
#pragma unroll
        for (int t = 0; t < COLT; ++t) {
            v2f b;
            b.x = w0[t * 16];
            b.y = w1[t * 16];
            acc[t] = __builtin_amdgcn_wmma_f32_16x16x4_f32(
                         false, a, false, b, (short)0, acc[t], false, false);
        }
    }
#pragma unroll
    for (int t = 0; t < COLT; ++t) {
        float bv = bias[t * 16 + l16];
#pragma unroll
        for (int i = 0; i < 8; ++i) {         // C/D: vgpr i -> M=i (lanes 0-15), M=i+8 (16-31)
            C[(size_t)(m0 + i + 8 * half) * KOUT + t * 16 + l16] = acc[t][i] + bv;
        }
    }
}

// ---------------------------------------------------------------------------
__global__ void fill_u32(unsigned* __restrict__ p, unsigned val, size_t n) {
    size_t i = (size_t)blockIdx.x * blockDim.x + threadIdx.x;
    size_t stride = (size_t)gridDim.x * blockDim.x;
    for (; i < n; i += stride) p[i] = val;
}

// ---------------------------------------------------------------------------
// CSR-by-destination build: histogram -> 3-kernel exclusive scan -> scatter.
// ---------------------------------------------------------------------------
__global__ void deg_count(const int* __restrict__ dst, int E, int* __restrict__ deg) {
    int e = blockIdx.x * blockDim.x + threadIdx.x;
    if (e < E) atomicAdd(&deg[dst[e]], 1);
}

__global__ void scan_block(const int* __restrict__ deg, int* __restrict__ excl,
                           int* __restrict__ bsum, int n) {
    __shared__ int sh[256];
    int i = blockIdx.x * 256 + threadIdx.x;
    int v = (i < n) ? deg[i] : 0;
    sh[threadIdx.x] = v;
    __syncthreads();
    for (int off = 1; off < 256; off <<= 1) {
        int t = (threadIdx.x >= off) ? sh[threadIdx.x - off] : 0;
        __syncthreads();
        sh[threadIdx.x] += t;
        __syncthreads();
    }
    if (i < n) excl[i] = sh[threadIdx.x] - v;            // exclusive scan
    if (threadIdx.x == 255) bsum[blockIdx.x] = sh[255];  // block total
}

__global__ void scan_tops(int* __restrict__ bsum, int nb) {  // nb <= 256, single block
    __shared__ int sh[256];
    int v = (threadIdx.x < nb) ? bsum[threadIdx.x] : 0;
    sh[threadIdx.x] = v;
    __syncthreads();
    for (int off = 1; off < 256; off <<= 1) {
        int t = (threadIdx.x >= off) ? sh[threadIdx.x - off] : 0;
        __syncthreads();
        sh[threadIdx.x] += t;
        __syncthreads();
    }
    if (threadIdx.x < nb) bsum[threadIdx.x] = sh[threadIdx.x] - v;
}

__global__ void scan_add(int* __restrict__ excl, const int* __restrict__ bsum, int n) {
    int i = blockIdx.x * blockDim.x + threadIdx.x;
    if (i < n) excl[i] += bsum[i >> 8];
}

__global__ void csr_scatter(const int* __restrict__ src, const int* __restrict__ dst,
                            int E, const int* __restrict__ rowstart,
                            int* __restrict__ cursor, int* __restrict__ srcs) {
    int e = blockIdx.x * blockDim.x + threadIdx.x;
    if (e >= E) return;
    int d = dst[e];
    int pos = rowstart[d] + atomicAdd(&cursor[d], 1);
    srcs[pos] = src[e];
}

// ---------------------------------------------------------------------------
// One wave per destination node: online-softmax attention over its CSR edge
// list. q-row pinned in registers; single pass reads k-row + v-row per edge;
// running max/sum/acc in registers; one normalized write. Zero atomics.
// Lane l owns channels [l*(D/32), ...): lanes 0-15 -> head0, 16-31 -> head1.
// ---------------------------------------------------------------------------
template <int D, int CPH>
__global__ void node_attention(const float* __restrict__ q,
                               const float* __restrict__ k,
                               const float* __restrict__ v,
                               const int* __restrict__ rowstart,
                               const int* __restrict__ deg,
                               const int* __restrict__ srcs,
                               float* __restrict__ agg, int n, float scale) {
    int node = (blockIdx.x * blockDim.x + threadIdx.x) >> 5;
    if (node >= n) return;
    int lane = threadIdx.x & 31;
    constexpr int Wc = D / 32;
    int c0 = lane * Wc;

    float qreg[Wc];
#pragma unroll
    for (int j = 0; j < Wc; ++j) qreg[j] = q[(size_t)node * D + c0 + j];

    int start = rowstart[node];
    int dg    = deg[node];

    float m = -__builtin_inff();
    float ssum = 0.f;
    float acc[Wc];
#pragma unroll
    for (int j = 0; j < Wc; ++j) acc[j] = 0.f;

    for (int i = 0; i < dg; ++i) {
        int s = srcs[start + i];
        const float* __restrict__ kr = k + (size_t)s * D + c0;
        const float* __restrict__ vr = v + (size_t)s * D + c0;
        float part = 0.f;
#pragma unroll
        for (int j = 0; j < Wc; ++j) part += qreg[j] * kr[j];
        // butterfly over each 16-lane half: every lane gets its head's dot
#pragma unroll
        for (int mm = 1; mm < 16; mm <<= 1) part += __shfl_xor(part, mm, 32);
        float sc = part * scale;
        if (sc > m) {                          // rescale running state
            float r = __expf(m - sc);
            ssum *= r;
#pragma unroll
            for (int j = 0; j < Wc; ++j) acc[j] *= r;
            m = sc;
        }
        float ex = __expf(sc - m);
        ssum += ex;
#pragma unroll
        for (int j = 0; j < Wc; ++j) acc[j] += ex * vr[j];
    }
    float inv = 1.f / ((ssum > 0.f) ? ssum : 1.f);
#pragma unroll
    for (int j = 0; j < Wc; ++j) agg[(size_t)node * D + c0 + j] = acc[j] * inv;
}

// ---------------------------------------------------------------------------
// Layer-1 node finalize: beta gate + LayerNorm + ReLU. Block of 128 = one node.
// ---------------------------------------------------------------------------
__global__ void node_final1(const float* __restrict__ agg,
                            const float* __restrict__ skip,
                            const float* __restrict__ Wb,
                            const float* __restrict__ g,
                            const float* __restrict__ b,
                            float* __restrict__ outp) {
    __shared__ float red[128];
    int node = blockIdx.x;
    int c = threadIdx.x;
    float ag = agg[(size_t)node * 128 + c];
    float sk = skip[(size_t)node * 128 + c];
    red[c] = ag * Wb[c] + sk * Wb[128 + c] + (ag - sk) * Wb[256 + c];
    __syncthreads();
    for (int s = 64; s > 0; s >>= 1) {
        if (c < s) red[c] += red[c + s];
        __syncthreads();
    }
    float beta = sigmoidf_(red[0]);
    __syncthreads();
    float y = beta * sk + (1.f - beta) * ag;
    red[c] = y;
    __syncthreads();
    for (int s = 64; s > 0; s >>= 1) {
        if (c < s) red[c] += red[c + s];
        __syncthreads();
    }
    float mu = red[0] * (1.f / 128.f);
    __syncthreads();
    float dm = y - mu;
    red[c] = dm * dm;
    __syncthreads();
    for (int s = 64; s > 0; s >>= 1) {
        if (c < s) red[c] += red[c + s];
        __syncthreads();
    }
    float var = red[0] * (1.f / 128.f);
    float hv = dm * rsqrtf(var + EPS_LN) * g[c] + b[c];
    outp[(size_t)node * 128 + c] = hv > 0.f ? hv : 0.f;
}

// ---------------------------------------------------------------------------
// Layer-2 node finalize: head mean + beta gate. One wave (32 lanes) per node.
// ---------------------------------------------------------------------------
__global__ void node_final2(const float* __restrict__ agg2,
                            const float* __restrict__ skip2,
                            const float* __restrict__ Wb2,
                            float* __restrict__ outp, int n) {
    int node = (blockIdx.x * blockDim.x + threadIdx.x) >> 5;
    if (node >= n) return;
    int c = threadIdx.x & 31;
    float am = 0.5f * (agg2[(size_t)node * 64 + c] + agg2[(size_t)node * 64 + 32 + c]);
    float sk = skip2[(size_t)node * 32 + c];
    float p = am * Wb2[c] + sk * Wb2[32 + c] + (am - sk) * Wb2[64 + c];
#pragma unroll
    for (int mm = 1; mm < 32; mm <<= 1) p += __shfl_xor(p, mm, 32);
    float beta = sigmoidf_(p);
    outp[(size_t)node * 32 + c] = beta * sk + (1.f - beta) * am;
}

// ---------------------------------------------------------------------------
extern "C" void kernel_launch(void* const* d_in, const int* in_sizes, int n_in,
                              void* d_out, int out_size, void* d_ws, size_t ws_size,
                              hipStream_t stream) {
    const float* x    = (const float*)d_in[0];
    const int*   ei   = (const int*)d_in[1];
    const float* Wq1  = (const float*)d_in[2];  const float* bq1 = (const float*)d_in[3];
    const float* Wk1  = (const float*)d_in[4];  const float* bk1 = (const float*)d_in[5];
    const float* Wv1  = (const float*)d_in[6];  const float* bv1 = (const float*)d_in[7];
    const float* Ws1  = (const float*)d_in[8];  const float* bs1 = (const float*)d_in[9];
    const float* Wb1  = (const float*)d_in[10];
    const float* lng1 = (const float*)d_in[11]; const float* lnb1 = (const float*)d_in[12];
    const float* Wq2  = (const float*)d_in[13]; const float* bq2 = (const float*)d_in[14];
    const float* Wk2  = (const float*)d_in[15]; const float* bk2 = (const float*)d_in[16];
    const float* Wv2  = (const float*)d_in[17]; const float* bv2 = (const float*)d_in[18];
    const float* Ws2  = (const float*)d_in[19]; const float* bs2 = (const float*)d_in[20];
    const float* Wb2  = (const float*)d_in[21];
    float* outp = (float*)d_out;

    const int N = in_sizes[0] / 128;
    const int E = in_sizes[1] / 2;
    const int* src = ei;
    const int* dst = ei + E;

    // ---- workspace layout ----
    float* Wf = (float*)d_ws;
    const size_t NB = (size_t)N * 128;
    float* q1   = Wf;
    float* k1   = Wf + NB;
    float* v1   = Wf + 2 * NB;
    float* s1   = Wf + 3 * NB;
    float* agg1 = Wf + 4 * NB;
    float* hbuf = Wf + 5 * NB;
    int*   ibuf     = (int*)(Wf + 6 * NB);
    int*   deg      = ibuf;                 // N
    int*   rowstart = ibuf + N;             // N
    int*   cursor   = ibuf + 2 * N;         // N
    int*   bsum     = ibuf + 3 * N;         // 256
    int*   srcs     = ibuf + 3 * N + 256;   // E
    // layer-2 buffers alias dead layer-1 buffers
    float* q2   = q1;
    float* k2   = q1 + (size_t)N * 64;
    float* v2   = k1;
    float* s2   = k1 + (size_t)N * 64;
    float* agg2 = v1;

    const int rowTiles = N / 16;                       // 3125
    const int gemmBlk  = (rowTiles * 32 + 255) / 256;  // 8 waves / block
    const int eBlk     = (E + 255) / 256;
    const int nBlk     = (N + 255) / 256;
    const int nb       = (N + 255) / 256;              // scan top count (<=256)
    const int waveNBlk = (N * 32 + 255) / 256;         // one wave per node

    // ---- CSR by destination (shared by both layers) ----
    fill_u32<<<256, 256, 0, stream>>>((unsigned*)deg, 0u, (size_t)N);
    fill_u32<<<256, 256, 0, stream>>>((unsigned*)cursor, 0u, (size_t)N);
    deg_count<<<eBlk, 256, 0, stream>>>(dst, E, deg);
    scan_block<<<nb, 256, 0, stream>>>(deg, rowstart, bsum, N);
    scan_tops<<<1, 256, 0, stream>>>(bsum, nb);
    scan_add<<<nBlk, 256, 0, stream>>>(rowstart, bsum, N);
    csr_scatter<<<eBlk, 256, 0, stream>>>(src, dst, E, rowstart, cursor, srcs);

    // ================= Layer 1 =================
    gemm_wmma_f32<128, 8><<<gemmBlk, 256, 0, stream>>>(x, Wq1, bq1, q1, rowTiles);
    gemm_wmma_f32<128, 8><<<gemmBlk, 256, 0, stream>>>(x, Wk1, bk1, k1, rowTiles);
    gemm_wmma_f32<128, 8><<<gemmBlk, 256, 0, stream>>>(x, Wv1, bv1, v1, rowTiles);
    gemm_wmma_f32<128, 8><<<gemmBlk, 256, 0, stream>>>(x, Ws1, bs1, s1, rowTiles);

    node_attention<128, 64><<<waveNBlk, 256, 0, stream>>>(
        q1, k1, v1, rowstart, deg, srcs, agg1, N, 0.125f /* 1/sqrt(64) */);

    node_final1<<<N, 128, 0, stream>>>(agg1, s1, Wb1, lng1, lnb1, hbuf);

    // ================= Layer 2 =================
    gemm_wmma_f32<128, 4><<<gemmBlk, 256, 0, stream>>>(hbuf, Wq2, bq2, q2, rowTiles);
    gemm_wmma_f32<128, 4><<<gemmBlk, 256, 0, stream>>>(hbuf, Wk2, bk2, k2, rowTiles);
    gemm_wmma_f32<128, 4><<<gemmBlk, 256, 0, stream>>>(hbuf, Wv2, bv2, v2, rowTiles);
    gemm_wmma_f32<128, 2><<<gemmBlk, 256, 0, stream>>>(hbuf, Ws2, bs2, s2, rowTiles);

    node_attention<64, 32><<<waveNBlk, 256, 0, stream>>>(
        q2, k2, v2, rowstart, deg, srcs, agg2, N, 0.17677669529663687f /* 1/sqrt(32) */);

    node_final2<<<(N * 32 + 255) / 256, 256, 0, stream>>>(agg2, s2, Wb2, outp, N);
}